// MaxTimesPlusErosionLiftingP4_76390288326908
// MI455X (gfx1250) — compile-verified
//
#include <hip/hip_runtime.h>

// ---------------- problem constants (from reference) ----------------
#define KH   7
#define PP   49            // KH*KH patch positions
#define CC   3             // channels
#define FF   32            // filters == wave32 lanes
#define HH   64
#define WW   64
#define BB   4
#define RR   4             // group rotations
#define EPSF 1e-7f
#define TAB  (PP*CC*FF)    // 4704 (inv,a) PAIRS per rotation => 2*TAB floats
#define TILE 16
#define HALO (TILE + KH - 1)   // 22

typedef unsigned int       u32;
typedef unsigned long long u64;
typedef __attribute__((ext_vector_type(4))) u32 u32x4_t;
typedef __attribute__((ext_vector_type(4))) int i32x4_t;
typedef __attribute__((ext_vector_type(8))) int i32x8_t;

#if defined(__gfx1250__) && __has_builtin(__builtin_amdgcn_tensor_load_to_lds)
#define HAVE_TDM 1
#else
#define HAVE_TDM 0
#endif

// ------------------------------------------------------------------
// TDM: 1D tile (nElem fp32, contiguous) from global -> LDS.
// D# per CDNA5 ISA ch.8 (group0: count|lds_addr|global_addr|type=2,
// group1: data_size=4B, dims/tile/stride).
// ------------------------------------------------------------------
#if HAVE_TDM
__device__ __forceinline__ void tdm_load_1d(const void* gsrc, void* ldsDst, u32 nElem) {
    u64 ga  = (u64)gsrc;
    u32 lds = (u32)(u64)ldsDst;            // low 32 bits of flat ptr = LDS offset
    u32x4_t g0;
    g0.x = 1u;                                              // count=1, user mode
    g0.y = lds;                                             // lds_addr
    g0.z = (u32)ga;                                         // global_addr[31:0]
    g0.w = (u32)((ga >> 32) & 0x1FFFFFFu) | 0x80000000u;    // ga[56:32] | type=2
    i32x8_t g1;
    g1[0] = (int)(2u << 16);          // workgroup_mask=0, data_size=2 (4B)
    g1[1] = (int)(nElem << 16);       // tensor_dim0[15:0] @ bits 63:48
    g1[2] = (int)(((nElem >> 16) & 0xFFFFu) | (1u << 16)); // dim0[31:16], tensor_dim1=1
    g1[3] = (int)(nElem << 16);       // tile_dim0 (<=65535), dim1[31:16]=0
    g1[4] = 1;                        // tile_dim1=1, tile_dim2=0
    g1[5] = (int)nElem;               // tensor_dim0_stride[31:0]
    g1[6] = 0;                        // stride0[47:32]=0, stride1[15:0]=0
    g1[7] = 0;
    i32x4_t z4 = (i32x4_t)0;
#if __clang_major__ >= 23
    i32x8_t z8 = (i32x8_t)0;
    __builtin_amdgcn_tensor_load_to_lds(g0, g1, z4, z4, z8, 0);
#else
    __builtin_amdgcn_tensor_load_to_lds(g0, g1, z4, z4, 0);
#endif
}
#endif

// ------------------------------------------------------------------
// Main kernel (placed FIRST so the disasm snippet shows the TDM issue).
// grid  = (16 tiles, 4 rotations, 4 batch), block = (32, 8): lane=f,
// threadIdx.y = wave; each wave owns 2 rows x 16 cols = 32 pixels.
// Table layout in ws/LDS: interleaved pairs tab2[(p*C+c)*F + f] = {inv, a}
// so the hot loop fetches both with one ds_load_b64.
// ------------------------------------------------------------------
__global__ __launch_bounds__(256)
void erosion_main(const float* __restrict__ x,
                  const float* __restrict__ tab,
                  float* __restrict__ out) {
    __shared__ __align__(16) float sTab[2 * TAB];         // (inv,a) pairs (37632 B)
    __shared__ __align__(16) float sX[HALO * HALO * CC];  // halo tile   ( 5808 B)

    const int rot = blockIdx.y;
    const int b   = blockIdx.z;
    const int tY  = (blockIdx.x >> 2) * TILE;
    const int tX  = (blockIdx.x & 3) * TILE;
    const int tid = threadIdx.y * 32 + threadIdx.x;
    const float* __restrict__ tabRot = tab + (size_t)rot * 2 * TAB;

    // ---- stage per-rotation (inv,a) table: one TDM async-tensor op ----
#if HAVE_TDM
    if (threadIdx.y == 0) {
        tdm_load_1d(tabRot, sTab, 2 * TAB);               // wave 0 issues TDM
    }
#else
    for (int j = tid; j < 2 * TAB; j += 256) sTab[j] = tabRot[j];
#endif

    // gfx1250 prefetch of the x halo rows (global_prefetch_b8)
    {
        int ry = tY - 3 + tid;
        if (tid < HALO && ry >= 0 && ry < HH) {
            int cx = (tX >= 3) ? (tX - 3) : 0;
            __builtin_prefetch(&x[(((size_t)b * HH + ry) * WW + cx) * CC], 0, 1);
        }
    }

    // ---- stage x tile with zero halo (SAME padding => zeros) ----
    for (int j = tid; j < HALO * HALO * CC; j += 256) {
        int c  = j % CC;
        int t  = j / CC;
        int xx = t % HALO;
        int yy = t / HALO;
        int gy = tY + yy - 3, gx = tX + xx - 3;
        float v = 0.0f;
        if (gy >= 0 && gy < HH && gx >= 0 && gx < WW)
            v = x[(((size_t)b * HH + gy) * WW + gx) * CC + c];
        sX[j] = v;
    }

#if HAVE_TDM
    if (threadIdx.y == 0) {
#if __has_builtin(__builtin_amdgcn_s_wait_tensorcnt)
        __builtin_amdgcn_s_wait_tensorcnt(0);
#else
        asm volatile("s_wait_tensorcnt 0x0" ::: "memory");
#endif
        asm volatile("" ::: "memory");   // TDM wrote sTab behind the compiler's back
    }
#endif
    __syncthreads();

    const int lane = threadIdx.x;                 // filter f
    const int wy   = threadIdx.y;                 // rows 2*wy, 2*wy+1
    const float2* __restrict__ sTab2 = (const float2*)sTab;

    float acc[32];
#pragma unroll
    for (int q = 0; q < 32; ++q) acc[q] = 0.0f;

    for (int c = 0; c < CC; ++c) {
        float m[32];
#pragma unroll
        for (int q = 0; q < 32; ++q) m[q] = __builtin_inff();
        for (int pr = 0; pr < KH; ++pr) {
            const int rbase = (2 * wy + pr) * (HALO * CC) + c;
            for (int pc = 0; pc < KH; ++pc) {
                const int p = pr * KH + pc;
                // one ds_load_b64: {inv, a} for (p,c,lane)
                const float2 t2 = sTab2[(p * CC + c) * FF + lane];
                const int base  = rbase + pc * CC;
#pragma unroll
                for (int q = 0; q < 32; ++q) {
                    // y is lane-uniform -> LDS broadcast read
                    float yv = sX[base + (q >> 4) * (HALO * CC) + (q & 15) * CC];
                    m[q] = fminf(m[q], fmaf(yv, t2.x, t2.y));
                }
            }
        }
#pragma unroll
        for (int q = 0; q < 32; ++q) acc[q] += m[q];
    }

    // out: [B,4,H,W,F], lane-contiguous stores (128B per wave per pixel)
#pragma unroll
    for (int q = 0; q < 32; ++q) {
        int gy = tY + 2 * wy + (q >> 4);
        int gx = tX + (q & 15);
        out[(((size_t)(b * RR + rot) * HH + gy) * WW + gx) * FF + lane] = acc[q];
    }
}

// ------------------------------------------------------------------
// rotation source index: for rotation i, output (r,c) reads original
// kernel[sr][sc].  (erosion base = rot180, then rot90^i, derived from
// jnp.rot90 semantics: rot1(m)[r,c] = m[c, N-1-r])
// ------------------------------------------------------------------
__device__ __forceinline__ void rot_src(int i, int r, int c, int& sr, int& sc) {
    switch (i & 3) {
        case 0:  sr = 6 - r; sc = 6 - c; break;   // rot180
        case 1:  sr = 6 - c; sc = r;     break;
        case 2:  sr = r;     sc = c;     break;
        default: sr = c;     sc = 6 - r; break;
    }
}

// ------------------------------------------------------------------
// Prep: interleaved-pair layout per rotation i:
//   tab[i*2*TAB + 2*((p*C+c)*F+f) + 0] = 1/(t_rot+eps)
//   tab[i*2*TAB + 2*((p*C+c)*F+f) + 1] = -k_rot/(t_rot+eps)
// ------------------------------------------------------------------
__global__ void prep_tables(const float* __restrict__ kern,
                            const float* __restrict__ times,
                            float* __restrict__ tab) {
    int idx = blockIdx.x * blockDim.x + threadIdx.x;
    if (idx >= RR * TAB) return;
    int i = idx / TAB;
    int o = idx - i * TAB;          // (p*C + c)*F + f
    int f = o & 31;
    int t = o >> 5;                 // p*C + c
    int c = t % CC;
    int p = t / CC;
    int r = p / KH, cc = p % KH;
    int sr, sc; rot_src(i, r, cc, sr, sc);
    int src = ((sr * KH + sc) * CC + c) * FF + f;
    float inv = 1.0f / (times[src] + EPSF);
    tab[i * 2 * TAB + 2 * o]     = inv;
    tab[i * 2 * TAB + 2 * o + 1] = -kern[src] * inv;
}

// ------------------------------------------------------------------
extern "C" void kernel_launch(void* const* d_in, const int* in_sizes, int n_in,
                              void* d_out, int out_size, void* d_ws, size_t ws_size,
                              hipStream_t stream) {
    const float* x     = (const float*)d_in[0];  // [4,64,64,3]
    const float* kern  = (const float*)d_in[1];  // [7,7,3,32]
    const float* times = (const float*)d_in[2];  // [7,7,3,32]
    float*       out   = (float*)d_out;          // [4,4,64,64,32]
    float*       tab   = (float*)d_ws;           // 8*TAB floats = 150528 B

    dim3 pb(256), pg((RR * TAB + 255) / 256);
    prep_tables<<<pg, pb, 0, stream>>>(kern, times, tab);

    dim3 mb(32, 8);
    dim3 mg((HH / TILE) * (WW / TILE), RR, BB);  // (16, 4, 4)
    erosion_main<<<mg, mb, 0, stream>>>(x, tab, out);
}